// GraphConv_59407987638624
// MI455X (gfx1250) — compile-verified
//
#include <hip/hip_runtime.h>

typedef __attribute__((ext_vector_type(2))) float v2f;
typedef __attribute__((ext_vector_type(8))) float v8f;

#define N_NODES   50000
#define N_EDGES   800000
#define DIM       64
#define N_LAYERS  4
#define N_CLASSES 10
#define N_GRAPHS  128

// ---------------------------------------------------------------------------
// Edge aggregation: agg[dst] += h[src].  One lane per (edge, feature-pair).
// 800000 edges * 32 lanes; each lane moves a float2 (8B) -> coalesced-ish
// gather + global_atomic_add_f32 scatter.  This is the bandwidth-bound core.
// ---------------------------------------------------------------------------
__global__ __launch_bounds__(256) void scatter_agg_kernel(
    const float* __restrict__ h, const int* __restrict__ src,
    const int* __restrict__ dst, float* __restrict__ agg)
{
    int tid = blockIdx.x * blockDim.x + threadIdx.x;   // < 25.6M
    int e = tid >> 5;
    if (e >= N_EDGES) return;
    int f = (tid & 31) * 2;
    int s = src[e];
    int d = dst[e];
    v2f v = *(const v2f*)(h + s * DIM + f);
    atomicAdd(&agg[d * DIM + f],     v.x);
    atomicAdd(&agg[d * DIM + f + 1], v.y);
}

// ---------------------------------------------------------------------------
// Fused layer GEMM: h_out = relu(agg @ W_rel + b_rel + h_in @ W_root)
// One wave per 16x16 output tile, K=64 via 16 steps of V_WMMA_F32_16X16X4_F32
// for each of the two GEMMs (32 chained WMMAs, accumulating in the same C).
// Exactly 12500 tiles = 3125 blocks * 4 waves; EXEC is all-ones everywhere.
// ---------------------------------------------------------------------------
__global__ __launch_bounds__(128) void gemm_layer_kernel(
    const float* __restrict__ agg, const float* __restrict__ hin,
    const float* __restrict__ Wrel, const float* __restrict__ brel,
    const float* __restrict__ Wroot, float* __restrict__ hout)
{
    int wave   = (blockIdx.x * 128 + threadIdx.x) >> 5;  // 0..12499
    int lane   = threadIdx.x & 31;
    int tile_n = wave & 3;          // 4 column tiles of 16 features
    int tile_m = wave >> 2;         // 3125 row tiles of 16 nodes
    int half   = lane >> 4;         // 0: lanes 0-15, 1: lanes 16-31
    int m      = lane & 15;         // A-matrix row within tile
    int n      = lane & 15;         // B/C/D column within tile
    int kb     = half << 1;         // K sub-offset: 0 or 2
    int col0   = tile_n * 16;

    const float* aggRow = agg + (tile_m * 16 + m) * DIM + kb;
    const float* hRow   = hin + (tile_m * 16 + m) * DIM + kb;

    v8f c = {};
#pragma unroll
    for (int s = 0; s < 16; ++s) {
        int k = s * 4;
        // agg @ W_rel contribution
        v2f a1 = *(const v2f*)(aggRow + k);
        v2f b1;
        b1.x = Wrel[(k + kb)     * DIM + col0 + n];
        b1.y = Wrel[(k + kb + 1) * DIM + col0 + n];
        c = __builtin_amdgcn_wmma_f32_16x16x4_f32(
                false, a1, false, b1, (short)0, c, false, false);
        // h @ W_root contribution
        v2f a2 = *(const v2f*)(hRow + k);
        v2f b2;
        b2.x = Wroot[(k + kb)     * DIM + col0 + n];
        b2.y = Wroot[(k + kb + 1) * DIM + col0 + n];
        c = __builtin_amdgcn_wmma_f32_16x16x4_f32(
                false, a2, false, b2, (short)0, c, false, false);
    }

    float bias = brel[col0 + n];
#pragma unroll
    for (int r = 0; r < 8; ++r) {
        int mr = r + (half << 3);               // C/D layout: VGPR r -> M=r / M=r+8
        float v = c[r] + bias;
        v = v > 0.0f ? v : 0.0f;                // relu
        hout[(tile_m * 16 + mr) * DIM + col0 + n] = v;
    }
}

// ---------------------------------------------------------------------------
// Global mean pool (sum + count phase): one lane per (node, feature-pair).
// ---------------------------------------------------------------------------
__global__ __launch_bounds__(256) void pool_kernel(
    const float* __restrict__ h, const int* __restrict__ batch,
    float* __restrict__ sums, float* __restrict__ counts)
{
    int tid = blockIdx.x * blockDim.x + threadIdx.x;   // < 1.6M
    int nid = tid >> 5;
    if (nid >= N_NODES) return;
    int lane = tid & 31;
    int f = lane * 2;
    int g = batch[nid];
    v2f v = *(const v2f*)(h + nid * DIM + f);
    atomicAdd(&sums[g * DIM + f],     v.x);
    atomicAdd(&sums[g * DIM + f + 1], v.y);
    if (lane == 0) atomicAdd(&counts[g], 1.0f);
}

// ---------------------------------------------------------------------------
// Final classifier: out[g,c] = (sums[g]/max(count,1)) @ W_lin + b_lin.
// 1280 outputs, trivial.
// ---------------------------------------------------------------------------
__global__ __launch_bounds__(256) void final_linear_kernel(
    const float* __restrict__ sums, const float* __restrict__ counts,
    const float* __restrict__ Wlin, const float* __restrict__ blin,
    float* __restrict__ out)
{
    int tid = blockIdx.x * blockDim.x + threadIdx.x;
    if (tid >= N_GRAPHS * N_CLASSES) return;
    int g = tid / N_CLASSES;
    int cls = tid % N_CLASSES;
    float cnt = counts[g];
    cnt = cnt > 1.0f ? cnt : 1.0f;
    float inv = 1.0f / cnt;
    float acc = blin[cls];
#pragma unroll 8
    for (int k = 0; k < DIM; ++k)
        acc += (sums[g * DIM + k] * inv) * Wlin[k * N_CLASSES + cls];
    out[tid] = acc;
}

extern "C" void kernel_launch(void* const* d_in, const int* in_sizes, int n_in,
                              void* d_out, int out_size, void* d_ws, size_t ws_size,
                              hipStream_t stream)
{
    const float* x      = (const float*)d_in[0];
    const int*   eidx   = (const int*)  d_in[1];
    const int*   batch  = (const int*)  d_in[2];
    const float* W_rel  = (const float*)d_in[3];
    const float* b_rel  = (const float*)d_in[4];
    const float* W_root = (const float*)d_in[5];
    const float* W_lin  = (const float*)d_in[6];
    const float* b_lin  = (const float*)d_in[7];

    const int* src = eidx;              // edge_index[0]
    const int* dst = eidx + N_EDGES;    // edge_index[1]

    char* ws = (char*)d_ws;
    const size_t featBytes = (size_t)N_NODES * DIM * sizeof(float);  // 12.8 MB
    float* agg    = (float*)(ws);
    float* hA     = (float*)(ws + featBytes);
    float* hB     = (float*)(ws + 2 * featBytes);
    float* sums   = (float*)(ws + 3 * featBytes);
    float* counts = sums + N_GRAPHS * DIM;

    const int scatterBlocks = (N_EDGES * 32 + 255) / 256;   // 100000
    const int gemmBlocks    = (N_NODES / 16) * 4 / 4;       // 12500 waves / 4 per block = 3125

    const float* hin  = x;
    float*       hout = hA;
    for (int l = 0; l < N_LAYERS; ++l) {
        hipMemsetAsync(agg, 0, featBytes, stream);
        scatter_agg_kernel<<<scatterBlocks, 256, 0, stream>>>(hin, src, dst, agg);
        gemm_layer_kernel<<<gemmBlocks, 128, 0, stream>>>(
            agg, hin,
            W_rel  + (size_t)l * DIM * DIM,
            b_rel  + (size_t)l * DIM,
            W_root + (size_t)l * DIM * DIM,
            hout);
        hin  = hout;
        hout = (hout == hA) ? hB : hA;
    }

    hipMemsetAsync(sums, 0, (size_t)(N_GRAPHS * DIM + N_GRAPHS) * sizeof(float), stream);
    pool_kernel<<<(N_NODES * 32 + 255) / 256, 256, 0, stream>>>(hin, batch, sums, counts);
    final_linear_kernel<<<(N_GRAPHS * N_CLASSES + 255) / 256, 256, 0, stream>>>(
        sums, counts, W_lin, b_lin, (float*)d_out);
}